// level_2_matrix_30502857736458
// MI455X (gfx1250) — compile-verified
//
#include <hip/hip_runtime.h>

typedef float v2f __attribute__((ext_vector_type(2)));
typedef float v8f __attribute__((ext_vector_type(8)));

constexpr int NF    = 39;     // fields
constexpr int D     = 16;     // embed dim
constexpr int BATCH = 8192;
constexpr int P     = 741;    // C(39,2)
constexpr int SLOTS = 750;    // pair slots incl. padded-range spill + dummy
constexpr int DUMMY = 749;    // clamp target for diagonal c>=r elements
constexpr int WPB   = 8;      // waves (= batch elements) per block
constexpr int TPB   = WPB * 32;
// LDS row stride 20 floats: 16-lane column access hits 16 distinct banks
// (20*row mod 64 all distinct for row=0..15) AND rows are 16B aligned so
// float4 staging stores and 8B-aligned v2f fragment loads are legal.
constexpr int XSTR  = 20;
constexpr float EPS = 1e-5f;

// ---------------------------------------------------------------------------
// Per-wave 48x48 Gram (lower-triangle tiles) of a 48x16 LDS-resident matrix
// using V_WMMA_F32_16X16X4_F32.  For X*X^T the B-fragment of row-block C has
// the same lane/VGPR layout as the A-fragment of row-block C, so 3 fragments
// per k-step feed all 6 lower-triangle tile WMMAs.
// acc index -> tile (R,C): 0:(0,0) 1:(1,0) 2:(1,1) 3:(2,0) 4:(2,1) 5:(2,2)
// ---------------------------------------------------------------------------
__device__ __forceinline__ void gram48(const float* __restrict__ Xs, int lane,
                                       v8f acc[6]) {
  const int m  = lane & 15;
  const int kh = (lane < 16) ? 0 : 2;
#pragma unroll
  for (int i = 0; i < 6; ++i) acc[i] = (v8f){0.f,0.f,0.f,0.f,0.f,0.f,0.f,0.f};
#pragma unroll
  for (int ks = 0; ks < 4; ++ks) {
    const int k0 = ks * 4 + kh;  // always even -> 8B-aligned LDS b64 loads
    const v2f f0 = *reinterpret_cast<const v2f*>(&Xs[( 0 + m) * XSTR + k0]);
    const v2f f1 = *reinterpret_cast<const v2f*>(&Xs[(16 + m) * XSTR + k0]);
    const v2f f2 = *reinterpret_cast<const v2f*>(&Xs[(32 + m) * XSTR + k0]);
    acc[0] = __builtin_amdgcn_wmma_f32_16x16x4_f32(false, f0, false, f0, (short)0, acc[0], false, false);
    acc[1] = __builtin_amdgcn_wmma_f32_16x16x4_f32(false, f1, false, f0, (short)0, acc[1], false, false);
    acc[2] = __builtin_amdgcn_wmma_f32_16x16x4_f32(false, f1, false, f1, (short)0, acc[2], false, false);
    acc[3] = __builtin_amdgcn_wmma_f32_16x16x4_f32(false, f2, false, f0, (short)0, acc[3], false, false);
    acc[4] = __builtin_amdgcn_wmma_f32_16x16x4_f32(false, f2, false, f1, (short)0, acc[4], false, false);
    acc[5] = __builtin_amdgcn_wmma_f32_16x16x4_f32(false, f2, false, f2, (short)0, acc[5], false, false);
  }
}

// pair index for (c < r), matching combinations(range(39), 2) order.
// Over the padded range (c < r <= 47) this stays within [0, 749].
__device__ __forceinline__ int pair_index(int c, int r) {
  return ((c * (77 - c)) >> 1) + r - c - 1;
}

// Stage one batch element's 39x16 block into padded 48x16 LDS tile (float4).
// Rows 39..47 are zero -> padded Gram entries are exactly 0.0f, so additions
// for r>=39 or c>=39 are numeric no-ops and need no validity checks.
__device__ __forceinline__ void load_x(const float* __restrict__ xb,
                                       float* __restrict__ Xs, int lane) {
  const float4* __restrict__ xb4 = reinterpret_cast<const float4*>(xb);
  const float4 zero = make_float4(0.f, 0.f, 0.f, 0.f);
  for (int idx = lane; idx < 48 * 4; idx += 32) {
    const int row = idx >> 2, q = idx & 3;
    const float4 v = (row < NF) ? xb4[row * 4 + q] : zero;
    *reinterpret_cast<float4*>(&Xs[row * XSTR + q * 4]) = v;  // 16B aligned
  }
}

// ---------------------------------------------------------------------------
// Kernel 0: zero S1/S2 accumulators (graph-replay safe)
// ---------------------------------------------------------------------------
__global__ void k_init(float* __restrict__ g) {
  const int i = blockIdx.x * blockDim.x + threadIdx.x;
  if (i < 2 * P) g[i] = 0.f;
}

// ---------------------------------------------------------------------------
// Kernel 1: WMMA Gram per batch element -> accumulate S1[p], S2[p].
// Off-diagonal tiles (R>C): c<r holds structurally -> fully unconditional
// ds_add_f32.  Diagonal tiles: single n < v+mhi compare, clamp to DUMMY.
// ---------------------------------------------------------------------------
__global__ __launch_bounds__(TPB)
void k_gram_stats(const float* __restrict__ x,
                  float* __restrict__ gS1, float* __restrict__ gS2) {
  __shared__ float Xs[WPB][48 * XSTR];
  __shared__ float S1s[SLOTS];
  __shared__ float S2s[SLOTS];
  const int tid  = threadIdx.x;
  const int lane = tid & 31;
  const int wave = tid >> 5;
  const int b    = blockIdx.x * WPB + wave;

  for (int i = tid; i < SLOTS; i += TPB) { S1s[i] = 0.f; S2s[i] = 0.f; }
  load_x(x + (size_t)b * (NF * D), &Xs[wave][0], lane);
  __syncthreads();

  v8f acc[6];
  gram48(&Xs[wave][0], lane, acc);

  const int n   = lane & 15;
  const int mhi = (lane < 16) ? 0 : 8;

  // off-diagonal tiles: acc idx 1:(1,0) 3:(2,0) 4:(2,1)
  const int OT[3]  = {1, 3, 4};
  const int OR[3]  = {16, 32, 32};
  const int OC[3]  = {0, 0, 16};
#pragma unroll
  for (int t = 0; t < 3; ++t) {
#pragma unroll
    for (int v = 0; v < 8; ++v) {
      const int r = OR[t] + v + mhi;
      const int c = OC[t] + n;
      const int p = pair_index(c, r);          // in [0,749]; padded adds +0.0
      const float val = acc[OT[t]][v];
      atomicAdd(&S1s[p], val);
      atomicAdd(&S2s[p], val * val);
    }
  }
  // diagonal tiles: acc idx 0:(0,0) 2:(1,1) 5:(2,2)
  const int DT[3] = {0, 2, 5};
  const int DB[3] = {0, 16, 32};
#pragma unroll
  for (int t = 0; t < 3; ++t) {
#pragma unroll
    for (int v = 0; v < 8; ++v) {
      const int r = DB[t] + v + mhi;
      const int c = DB[t] + n;
      const bool ok = n < (v + mhi);           // strict lower triangle
      const int p = ok ? pair_index(c, r) : DUMMY;
      const float val = ok ? acc[DT[t]][v] : 0.f;
      atomicAdd(&S1s[p], val);
      atomicAdd(&S2s[p], val * val);
    }
  }
  __syncthreads();
  for (int i = tid; i < P; i += TPB) {         // slots >= P never read back
    atomicAdd(&gS1[i], S1s[i]);
    atomicAdd(&gS2[i], S2s[i]);
  }
}

// ---------------------------------------------------------------------------
// Kernel 2: finalize batch-norm stats -> a[p], Csum
// ---------------------------------------------------------------------------
__global__ void k_stats(const float* __restrict__ gS1, const float* __restrict__ gS2,
                        const float* __restrict__ ew, const float* __restrict__ gamma,
                        const float* __restrict__ beta,
                        float* __restrict__ gA, float* __restrict__ gCsum) {
  __shared__ float cs[P];
  const int tid = threadIdx.x;
  const float invB = 1.0f / (float)BATCH;
  for (int p = tid; p < P; p += blockDim.x) {
    const float mean = gS1[p] * invB;
    const float var  = gS2[p] * invB - mean * mean;
    const float inv  = rsqrtf(var + EPS);
    const float a    = ew[p] * gamma[p] * inv;
    gA[p] = a;
    cs[p] = ew[p] * beta[p] - a * mean;
  }
  __syncthreads();
  if (tid == 0) {
    float s = 0.f;
    for (int p = 0; p < P; ++p) s += cs[p];
    gCsum[0] = s;
  }
}

// ---------------------------------------------------------------------------
// Kernel 3: WMMA Gram again, contract against a[p], reduce per wave -> out[b]
// Weight slots >= P are zero-filled so padded elements (val==0) can index
// them safely (avoids poison-LDS NaN * 0).
// ---------------------------------------------------------------------------
__global__ __launch_bounds__(TPB)
void k_output(const float* __restrict__ x, const float* __restrict__ gA,
              const float* __restrict__ gCsum, float* __restrict__ out) {
  __shared__ float Xs[WPB][48 * XSTR];
  __shared__ float as_[SLOTS];
  const int tid  = threadIdx.x;
  const int lane = tid & 31;
  const int wave = tid >> 5;
  const int b    = blockIdx.x * WPB + wave;

  for (int i = tid; i < SLOTS; i += TPB) as_[i] = (i < P) ? gA[i] : 0.f;
  load_x(x + (size_t)b * (NF * D), &Xs[wave][0], lane);
  __syncthreads();

  v8f acc[6];
  gram48(&Xs[wave][0], lane, acc);

  const int n   = lane & 15;
  const int mhi = (lane < 16) ? 0 : 8;
  float s = 0.f;

  const int OT[3]  = {1, 3, 4};
  const int OR[3]  = {16, 32, 32};
  const int OC[3]  = {0, 0, 16};
#pragma unroll
  for (int t = 0; t < 3; ++t) {
#pragma unroll
    for (int v = 0; v < 8; ++v) {
      const int r = OR[t] + v + mhi;
      const int c = OC[t] + n;
      s += as_[pair_index(c, r)] * acc[OT[t]][v];   // padded: 0 * x
    }
  }
  const int DT[3] = {0, 2, 5};
  const int DB[3] = {0, 16, 32};
#pragma unroll
  for (int t = 0; t < 3; ++t) {
#pragma unroll
    for (int v = 0; v < 8; ++v) {
      const int r = DB[t] + v + mhi;
      const int c = DB[t] + n;
      const bool ok = n < (v + mhi);
      const int p = ok ? pair_index(c, r) : DUMMY;  // as_[DUMMY] == 0
      s += as_[p] * acc[DT[t]][v];
    }
  }
  // wave32 tree reduction
#pragma unroll
  for (int off = 16; off > 0; off >>= 1) s += __shfl_xor(s, off, 32);
  if (lane == 0) out[b] = s + gCsum[0];
}

// ---------------------------------------------------------------------------
extern "C" void kernel_launch(void* const* d_in, const int* in_sizes, int n_in,
                              void* d_out, int out_size, void* d_ws, size_t ws_size,
                              hipStream_t stream) {
  const float* x     = (const float*)d_in[0];
  const float* ew    = (const float*)d_in[1];
  const float* gamma = (const float*)d_in[2];
  const float* beta  = (const float*)d_in[3];
  // d_in[4] (rows) / d_in[5] (cols) unused: pair index computed analytically
  float* out = (float*)d_out;
  float* w   = (float*)d_ws;
  float* gS1 = w;          // [P]
  float* gS2 = w + P;      // [P]
  float* gA  = w + 2 * P;  // [P]
  float* gC  = w + 3 * P;  // [1]

  k_init<<<(2 * P + 255) / 256, 256, 0, stream>>>(gS1);
  k_gram_stats<<<BATCH / WPB, TPB, 0, stream>>>(x, gS1, gS2);
  k_stats<<<1, 256, 0, stream>>>(gS1, gS2, ew, gamma, beta, gA, gC);
  k_output<<<BATCH / WPB, TPB, 0, stream>>>(x, gA, gC, out);
}